// DWTFeatureModel_7275674600053
// MI455X (gfx1250) — compile-verified
//
#include <hip/hip_runtime.h>
#include <hip/hip_bf16.h>

typedef __attribute__((ext_vector_type(2))) float v2f;
typedef __attribute__((ext_vector_type(8))) float v8f;

#define NB    2048   // batches
#define NF    64     // filters
#define TLEN  256    // time samples
#define TDWT  276    // DWT coeffs per signal
#define NPOS  64     // spatial positions (8x8)
#define BQ    16     // batches per workgroup (WMMA M tile)
#define GPOS  8      // positions staged in LDS per group
#define PPH   4      // positions per DWT/GEMM phase (4 threads per signal)
#define KGP   72     // K-groups of 4 per position (276 padded to 288)
#define KGH   36     // K-groups per wave-half (compile-time trip count)
#define SROW  294    // s_d row stride in floats (even for b64 align; 294%64=38 -> conflict-free)

__device__ __forceinline__ int refl_idx(int i, int N) {
    i = (i < 0) ? -i : i;
    i = (i >= N) ? (2 * N - 2 - i) : i;
    return i;
}

// db4 decomposition filters, pre-reversed so y[k] = sum_i xp[2k+i]*H[i].
// Sliding-window: consecutive k share 6 of 8 taps -> only 2 LDS reads per coeff-pair.
// 4 threads per signal (sub4 = 0..3), each owns a contiguous k-range.
__device__ __forceinline__ void dwt_level(
    const float* __restrict__ inb,   // LDS, element t at inb[t*16] (already offset by b)
    int N, int nout, int sub4,
    float* __restrict__ lo, int lostride,   // lo[k*lostride]
    float* __restrict__ hi)                 // hi[k]
{
    const float HLO[8] = { 0.23037781330885523f,  0.7148465705525415f,
                           0.6308807679295904f,  -0.02798376941698385f,
                          -0.18703481171888114f,  0.030841381835986965f,
                           0.032883011666982945f,-0.010597401784997278f};
    const float HHI[8] = {-0.010597401784997278f,-0.032883011666982945f,
                           0.030841381835986965f, 0.18703481171888114f,
                          -0.02798376941698385f, -0.6308807679295904f,
                           0.7148465705525415f,  -0.23037781330885523f};
    int per = (nout + 3) >> 2;
    int k0 = sub4 * per;
    int k1 = k0 + per; if (k1 > nout) k1 = nout;
    if (k0 >= k1) return;

    // window for coeff k covers sig[2k-6 .. 2k+1] (reflect-padded)
    float w0, w1, w2, w3, w4, w5, w6, w7;
    {
        int base = 2 * k0 - 6;
        w0 = inb[refl_idx(base + 0, N) * 16];
        w1 = inb[refl_idx(base + 1, N) * 16];
        w2 = inb[refl_idx(base + 2, N) * 16];
        w3 = inb[refl_idx(base + 3, N) * 16];
        w4 = inb[refl_idx(base + 4, N) * 16];
        w5 = inb[refl_idx(base + 5, N) * 16];
        w6 = inb[refl_idx(base + 6, N) * 16];
        w7 = inb[refl_idx(base + 7, N) * 16];
    }
    for (int k = k0;;) {
        float l, h;
        l = w0 * HLO[0];             h = w0 * HHI[0];
        l = fmaf(w1, HLO[1], l);     h = fmaf(w1, HHI[1], h);
        l = fmaf(w2, HLO[2], l);     h = fmaf(w2, HHI[2], h);
        l = fmaf(w3, HLO[3], l);     h = fmaf(w3, HHI[3], h);
        l = fmaf(w4, HLO[4], l);     h = fmaf(w4, HHI[4], h);
        l = fmaf(w5, HLO[5], l);     h = fmaf(w5, HHI[5], h);
        l = fmaf(w6, HLO[6], l);     h = fmaf(w6, HHI[6], h);
        l = fmaf(w7, HLO[7], l);     h = fmaf(w7, HHI[7], h);
        lo[k * lostride] = l;
        hi[k] = h;
        if (++k >= k1) break;
        w0 = w2; w1 = w3; w2 = w4; w3 = w5; w4 = w6; w5 = w7;
        w6 = inb[refl_idx(2 * k,     N) * 16];
        w7 = inb[refl_idx(2 * k + 1, N) * 16];
    }
}

// Pre-swizzle W into WMMA B-fragment layout (K padded to 288 with zeros):
// wf[((nt*64 + p)*72 + kg)*32 + lane] = float2 of
//   W[f = nt*16+(lane&15)][t = kg*4+2*(lane>>4)+{0,1}][p], 0 if t >= 276
__global__ __launch_bounds__(256) void swizzleW_kernel(
    const float* __restrict__ W, float* __restrict__ wf)
{
    int i = blockIdx.x * blockDim.x + threadIdx.x;
    const int NT = 4 * NPOS * KGP * 32;
    if (i >= NT) return;
    int l  = i & 31;
    int kg = (i >> 5) % KGP;
    int p  = (i / (32 * KGP)) % NPOS;
    int nt = i / (32 * KGP * NPOS);
    int f  = nt * 16 + (l & 15);
    int t  = kg * 4 + ((l >> 4) << 1);
    v2f v;
    v.x = (t     < TDWT) ? W[((size_t)f * TDWT + t    ) * NPOS + p] : 0.f;
    v.y = (t + 1 < TDWT) ? W[((size_t)f * TDWT + t + 1) * NPOS + p] : 0.f;
    *(v2f*)(wf + (size_t)i * 2) = v;
}

__global__ __launch_bounds__(256) void dwt_wmma_fused_kernel(
    const float* __restrict__ x, const float* __restrict__ wf,
    const float* __restrict__ bias, float* __restrict__ out)
{
    // ~255 KB static LDS (CDNA5 WGP has 320 KB)
    __shared__ __align__(16) float s_sig[GPOS * TLEN * 16];   // [pl][t][b]      128 KB
    __shared__ __align__(16) float s_d[PPH * 16 * SROW];      // [pos][b][294]   75.3 KB
    __shared__ __align__(16) float s_lo1[PPH * 131 * 16];     // [pos][t][b]     33.5 KB
    __shared__ __align__(16) float s_lo2[PPH * 69 * 16];      // [pos][t][b]     17.7 KB
    __shared__ __align__(16) float s_red[4 * 8 * 32];         //                 4 KB

    const int tid  = threadIdx.x;
    const int wave = tid >> 5;
    const int lane = tid & 31;
    const int B0   = blockIdx.x * BQ;

    // GEMM wave roles: 4 N-tiles x 2 K-halves of 36 k-groups each
    const int nt   = wave & 3;
    const int half = wave >> 2;

    // DWT thread roles: 4 threads per signal, 4 positions in flight
    const int b     = tid & 15;
    const int sub   = tid >> 4;      // 0..15
    const int pos_l = sub >> 2;      // 0..3 : position within phase
    const int sub4  = sub & 3;       // 0..3 : k-range within signal

    // zero the padded K tail of s_d (t = 276..287 per row); DWT never writes it
    for (int i = tid; i < PPH * 16 * (KGP * 4 - TDWT); i += 256) {
        int row = i / (KGP * 4 - TDWT);
        int c   = i % (KGP * 4 - TDWT);
        s_d[row * SROW + TDWT + c] = 0.f;
    }

    v8f acc = {};

    const int arow = lane & 15;
    const int koff = (lane >> 4) << 1;

    for (int g = 0; g < NPOS / GPOS; ++g) {
        // ---- stage x for 8 positions: float4 along p (32B-contiguous chunks) ----
        for (int i = tid; i < 16 * TLEN * (GPOS / 4); i += 256) {
            int c  = i & (GPOS / 4 - 1);
            int t  = (i / (GPOS / 4)) & 255;
            int bb = i / ((GPOS / 4) * TLEN);
            float4 v = *(const float4*)(x + (size_t)(B0 + bb) * (TLEN * NPOS)
                                          + t * NPOS + g * GPOS + c * 4);
            int pl = c * 4;
            s_sig[((pl + 0) * TLEN + t) * 16 + bb] = v.x;
            s_sig[((pl + 1) * TLEN + t) * 16 + bb] = v.y;
            s_sig[((pl + 2) * TLEN + t) * 16 + bb] = v.z;
            s_sig[((pl + 3) * TLEN + t) * 16 + bb] = v.w;
        }
        __syncthreads();

        for (int fh = 0; fh < GPOS / PPH; ++fh) {
            const int pbase = g * GPOS + fh * PPH;

            // prefetch this phase's B-fragment stream for this wave (~12 KB of 36.8 KB)
            {
                const char* nb = (const char*)(wf
                    + (((size_t)(nt * NPOS + pbase) * KGP + (size_t)half * KGH) * 32) * 2);
                __builtin_prefetch(nb + (size_t)lane * 128,        0, 1);
                __builtin_prefetch(nb + 4096 + (size_t)lane * 128, 0, 1);
                __builtin_prefetch(nb + 8192 + (size_t)lane * 128, 0, 1);
            }

            // ---- DWT for 4 positions (one signal per (b, pos_l), 4 subs each) ----
            const float* sig = s_sig + (fh * PPH + pos_l) * (TLEN * 16) + b;
            float* drow = s_d + (pos_l * 16 + b) * SROW;
            const float* lo1 = s_lo1 + pos_l * (131 * 16) + b;
            const float* lo2 = s_lo2 + pos_l * (69 * 16) + b;
            // d layout per signal: [lo3 0..37 | hi1 38..168 | hi2 169..237 | hi3 238..275 | pad]
            dwt_level(sig,        256, 131, sub4, (float*)lo1, 16, drow + 38);
            __syncthreads();
            dwt_level(lo1,        131,  69, sub4, (float*)lo2, 16, drow + 169);
            __syncthreads();
            dwt_level(lo2,         69,  38, sub4, drow, 1, drow + 238);
            __syncthreads();

            // ---- WMMA f32 16x16x4: 4 positions x 36 k-groups per wave ----
            for (int q = 0; q < PPH; ++q) {
                const v2f* wfp = (const v2f*)wf
                    + ((size_t)(nt * NPOS + pbase + q) * KGP + (size_t)half * KGH) * 32;
                const float* ap = s_d + (q * 16 + arow) * SROW + half * (KGH * 4) + koff;
#pragma unroll 12
                for (int j = 0; j < KGH; ++j) {
                    v2f a  = *(const v2f*)(ap + j * 4);
                    v2f bf = wfp[j * 32 + lane];
                    acc = __builtin_amdgcn_wmma_f32_16x16x4_f32(
                        false, a, false, bf, (short)0, acc, false, false);
                }
            }
            __syncthreads();
        }
    }

    // ---- reduce the two K-halves, bias + leaky ReLU, store ----
    if (half == 1) {
#pragma unroll
        for (int r = 0; r < 8; ++r) s_red[(nt * 8 + r) * 32 + lane] = acc[r];
    }
    __syncthreads();
    if (half == 0) {
        int n   = lane & 15;
        int mhi = (lane >> 4) * 8;   // C/D layout: VGPR r -> M = r (lanes 0-15), r+8 (lanes 16-31)
        int f   = nt * 16 + n;
        float bv = bias[f];
#pragma unroll
        for (int r = 0; r < 8; ++r) {
            float v = acc[r] + s_red[(nt * 8 + r) * 32 + lane] + bv;
            v = (v >= 0.f) ? v : 0.02f * v;
            out[(size_t)(B0 + mhi + r) * NF + f] = v;
        }
    }
}

extern "C" void kernel_launch(void* const* d_in, const int* in_sizes, int n_in,
                              void* d_out, int out_size, void* d_ws, size_t ws_size,
                              hipStream_t stream) {
    const float* x    = (const float*)d_in[0];
    const float* W    = (const float*)d_in[1];
    const float* bias = (const float*)d_in[2];
    float* out = (float*)d_out;
    float* wf  = (float*)d_ws;   // needs 4*64*72*32*2 floats = ~4.7 MB

    const int nswz = 4 * NPOS * KGP * 32;
    swizzleW_kernel<<<(nswz + 255) / 256, 256, 0, stream>>>(W, wf);
    dwt_wmma_fused_kernel<<<NB / BQ, 256, 0, stream>>>(x, wf, bias, out);
}